// Decoder_predict_36782099923051
// MI455X (gfx1250) — compile-verified
//
#include <hip/hip_runtime.h>
#include <stdint.h>

#define B      256
#define N      4096
#define T      30
#define KSEL   6
#define BLK    256
#define PT     (N / BLK)      // 16 candidates per thread (strided)
#define NW     (BLK / 32)     // 8 wave32s per block
#define THR2   4.0f

__global__ __launch_bounds__(BLK) void nms_goals_kernel(
    const float* __restrict__ coord,      // [B,1,N,2]
    const float* __restrict__ cls,        // [B,1,N]
    const float* __restrict__ traj,       // [B,1,N,T,2]
    const float* __restrict__ cen,        // [B,1,N]
    float* __restrict__ out)              // trajs | probs | goals
{
    __shared__ __align__(16) float sCxy[2 * N];       // 32 KB async staging
    __shared__ float redS[NW];
    __shared__ int   redI[NW];
    __shared__ float wS;                              // per-pass winner broadcast
    __shared__ int   wI;
    __shared__ float selP[KSEL], selX[KSEL], selY[KSEL];
    __shared__ int   selI[KSEL];

    const int b    = blockIdx.x;
    const int tid  = threadIdx.x;
    const int lane = tid & 31;
    const int wave = tid >> 5;
    const float NEGINF = -__builtin_inff();

    const float* clsB = cls + (size_t)b * N;
    const float* cenB = cen + (size_t)b * N;
    const float* crdB = coord + (size_t)b * N * 2;

    // ---- Stage coords via CDNA5 async global->LDS DMA (b128, ASYNCcnt) ----
    #pragma unroll
    for (int j = 0; j < 8; ++j) {
        int e = j * BLK + tid;                         // float4 index
        uint32_t ldsb  = (uint32_t)(uintptr_t)(const void*)&sCxy[e * 4];
        uint64_t gaddr = (uint64_t)(uintptr_t)(crdB + (size_t)e * 4);
        asm volatile("global_load_async_to_lds_b128 %0, %1, off"
                     :: "v"(ldsb), "v"(gaddr) : "memory");
    }

    // ---- Scores straight into registers (coalesced, overlaps the DMA) ----
    float sc[PT];
    #pragma unroll
    for (int j = 0; j < PT; ++j) {
        int i = j * BLK + tid;
        sc[j] = clsB[i] * cenB[i];
    }

    asm volatile("s_wait_asynccnt 0" ::: "memory");
    __syncthreads();

    // ---- Coords LDS -> registers (b64 reads, all 64 banks, conflict-free) ----
    float cx[PT], cy[PT];
    const float2* c2 = (const float2*)sCxy;
    #pragma unroll
    for (int j = 0; j < PT; ++j) {
        float2 p = c2[j * BLK + tid];
        cx[j] = p.x; cy[j] = p.y;
    }

    // ---- Greedy NMS: 6 x (fused suppress + stable argmax), register-resident ----
    bool  lastValid = false;                          // pass 0: no suppression
    float pwx = 0.0f, pwy = 0.0f;
    #pragma unroll 1
    for (int k = 0; k < KSEL; ++k) {
        if (lastValid) {
            #pragma unroll
            for (int j = 0; j < PT; ++j) {
                float dx = cx[j] - pwx, dy = cy[j] - pwy;
                if (dx * dx + dy * dy < THR2) sc[j] = NEGINF;
            }
        }
        // local stable argmax (ascending local index + strict >)
        float bs = NEGINF;
        int   bi = 0x7fffffff;
        #pragma unroll
        for (int j = 0; j < PT; ++j) {
            if (sc[j] > bs) { bs = sc[j]; bi = j * BLK + tid; }
        }
        // intra-wave (score, idx) reduction: 5 shuffle steps, no barriers
        #pragma unroll
        for (int off = 16; off > 0; off >>= 1) {
            float s2 = __shfl_down(bs, off, 32);
            int   i2 = __shfl_down(bi, off, 32);
            if (s2 > bs || (s2 == bs && i2 < bi)) { bs = s2; bi = i2; }
        }
        if (lane == 0) { redS[wave] = bs; redI[wave] = bi; }
        __syncthreads();
        if (wave == 0) {
            float rs = (lane < NW) ? redS[lane] : NEGINF;
            int   ri = (lane < NW) ? redI[lane] : 0x7fffffff;
            #pragma unroll
            for (int off = 4; off > 0; off >>= 1) {
                float s2 = __shfl_down(rs, off, 32);
                int   i2 = __shfl_down(ri, off, 32);
                if (s2 > rs || (s2 == rs && i2 < ri)) { rs = s2; ri = i2; }
            }
            if (lane == 0) {
                if (rs > NEGINF) {                     // accept
                    selP[k] = rs; selI[k] = ri;
                    selX[k] = sCxy[2 * ri]; selY[k] = sCxy[2 * ri + 1];
                } else {                               // reference default fill
                    selP[k] = selP[0]; selI[k] = selI[0];
                    selX[k] = 0.0f;    selY[k] = 0.0f;
                }
                wS = rs; wI = ri;
            }
        }
        __syncthreads();
        float ws = wS;
        int   wi = wI;
        lastValid = (ws > NEGINF);                     // block-uniform
        if (lastValid) {                               // broadcast LDS read
            pwx = sCxy[2 * wi];
            pwy = sCxy[2 * wi + 1];
        }
    }
    __syncthreads();                                   // sel* visible to all

    // ---- Outputs: trajs [B,K,T,2] | probs [B,K] | goals [B,K,2] ----
    float* outTraj = out;
    float* outProb = out + (size_t)B * KSEL * T * 2;
    float* outGoal = outProb + (size_t)B * KSEL;

    const float* trjB = traj + (size_t)b * N * T * 2;
    for (int o = tid; o < KSEL * T * 2; o += BLK) {    // 360 floats, coalesced
        int k = o / (T * 2);
        int e = o % (T * 2);
        outTraj[((size_t)b * KSEL + k) * (T * 2) + e] =
            trjB[(size_t)selI[k] * (T * 2) + e];
    }
    if (tid < KSEL)     outProb[(size_t)b * KSEL + tid] = selP[tid];
    if (tid < KSEL * 2) outGoal[(size_t)b * KSEL * 2 + tid] =
                            (tid & 1) ? selY[tid >> 1] : selX[tid >> 1];
}

extern "C" void kernel_launch(void* const* d_in, const int* in_sizes, int n_in,
                              void* d_out, int out_size, void* d_ws, size_t ws_size,
                              hipStream_t stream) {
    (void)in_sizes; (void)n_in; (void)d_ws; (void)ws_size; (void)out_size;
    const float* coord = (const float*)d_in[0];   // [B,1,N,2]
    const float* cls   = (const float*)d_in[1];   // [B,1,N]
    const float* traj  = (const float*)d_in[2];   // [B,1,N,T,2]
    const float* cen   = (const float*)d_in[3];   // [B,1,N]
    float* out = (float*)d_out;
    nms_goals_kernel<<<B, BLK, 0, stream>>>(coord, cls, traj, cen, out);
}